// RGBLoss_19172734009361
// MI455X (gfx1250) — compile-verified
//
#include <hip/hip_runtime.h>
#include <math.h>

typedef float v2f __attribute__((ext_vector_type(2)));
typedef float v8f __attribute__((ext_vector_type(8)));

#define H_IMG 240
#define W_IMG 320
#define FX 277.0f
#define FY 277.0f
#define CX 160.0f
#define CY 120.0f
#define N_STEPS 64
#define S_OUT 160
#define D_IN 96
#define CAM_SCALE 0.0301f
#define SC (95.0f / 159.0f)           // (D-1)/(out-1)
#define NRAYS (3 * H_IMG * W_IMG)     // 230400
#define RM_BLOCKS (NRAYS / 256)       // 900

// ---------- trilerp of the 96^3 source grid (coords in [0,95]) ----------
__device__ __forceinline__ float tri96(const float* __restrict__ g,
                                       float x, float y, float z) {
    x = fminf(x, 95.0f); y = fminf(y, 95.0f); z = fminf(z, 95.0f);
    int x0 = (int)x, y0 = (int)y, z0 = (int)z;
    float fx = x - (float)x0, fy = y - (float)y0, fz = z - (float)z0;
    int x1 = min(x0 + 1, 95), y1 = min(y0 + 1, 95), z1 = min(z0 + 1, 95);
    int bx0 = x0 * 96 * 96, bx1 = x1 * 96 * 96;
    int by0 = y0 * 96,      by1 = y1 * 96;
    float c000 = g[bx0 + by0 + z0], c100 = g[bx1 + by0 + z0];
    float c010 = g[bx0 + by1 + z0], c001 = g[bx0 + by0 + z1];
    float c110 = g[bx1 + by1 + z0], c101 = g[bx1 + by0 + z1];
    float c011 = g[bx0 + by1 + z1], c111 = g[bx1 + by1 + z1];
    float gx = 1.0f - fx, gy = 1.0f - fy, gz = 1.0f - fz;
    return c000*gx*gy*gz + c100*fx*gy*gz + c010*gx*fy*gz + c001*gx*gy*fz
         + c110*fx*fy*gz + c101*fx*gy*fz + c011*gx*fy*fz + c111*fx*fy*fz;
}

// ---------- trilerp of the materialized 160^3 SDF (reference clamp) ----------
__device__ __forceinline__ float tri_sdf(const float* __restrict__ v,
                                         float x, float y, float z) {
    x = fminf(fmaxf(x, 0.0f), 158.999f);
    y = fminf(fmaxf(y, 0.0f), 158.999f);
    z = fminf(fmaxf(z, 0.0f), 158.999f);
    int x0 = (int)x, y0 = (int)y, z0 = (int)z;
    float fx = x - (float)x0, fy = y - (float)y0, fz = z - (float)z0;
    int x1 = x0 + 1, y1 = y0 + 1, z1 = z0 + 1;  // <=159 by clamp
    int bx0 = x0 * S_OUT * S_OUT, bx1 = x1 * S_OUT * S_OUT;
    int by0 = y0 * S_OUT,         by1 = y1 * S_OUT;
    float c000 = v[bx0 + by0 + z0], c100 = v[bx1 + by0 + z0];
    float c010 = v[bx0 + by1 + z0], c001 = v[bx0 + by0 + z1];
    float c110 = v[bx1 + by1 + z0], c101 = v[bx1 + by0 + z1];
    float c011 = v[bx0 + by1 + z1], c111 = v[bx1 + by1 + z1];
    float gx = 1.0f - fx, gy = 1.0f - fy, gz = 1.0f - fz;
    return c000*gx*gy*gz + c100*fx*gy*gz + c010*gx*fy*gz + c001*gx*gy*fz
         + c110*fx*fy*gz + c101*fx*gy*fz + c011*gx*fy*fz + c111*fx*fy*fz;
}

// ===== Kernel 1: materialize sdf_rs = trilerp_resize(sdf96) - TRUNC =====
__global__ void k_resize_sdf(const float* __restrict__ sdf96,
                             float* __restrict__ sdf_rs) {
    int idx = blockIdx.x * blockDim.x + threadIdx.x;   // 160^3 = 4,096,000
    int k = idx % S_OUT;
    int j = (idx / S_OUT) % S_OUT;
    int i = idx / (S_OUT * S_OUT);
    float val = tri96(sdf96, (float)i * SC, (float)j * SC, (float)k * SC);
    sdf_rs[idx] = val - 1.5f;
}

// ===== Kernel 2: dirs = normalize(R . dcam) via v_wmma_f32_16x16x4_f32 =====
// One wave per 16-pixel strip. A = R (M=component, K=j), B = dcam^T (K=j, N=pixel).
// D[M=0..2, N=n] lands in lanes 0..15, VGPRs 0..2.
__global__ void k_dirs_wmma(const float* __restrict__ poses,
                            float* __restrict__ dirs) {
    int lane = threadIdx.x & 31;
    int wave = threadIdx.x >> 5;
    int strip = blockIdx.x * (blockDim.x >> 5) + wave;  // exactly 14400 strips
    int v    = strip / ((H_IMG * W_IMG) / 16);
    int rem  = strip % ((H_IMG * W_IMG) / 16);
    int pix0 = rem * 16;
    int h  = pix0 / W_IMG;
    int w0 = pix0 % W_IMG;

    const float* M = poses + v * 16;
    // A operand (16x4 f32): V0 = {K=0 | K=2}, V1 = {K=1 | K=3} across half-waves
    float a0, a1;
    if (lane < 16) {
        a0 = (lane < 3) ? M[lane * 4 + 0] : 0.0f;   // R[m][0]
        a1 = (lane < 3) ? M[lane * 4 + 1] : 0.0f;   // R[m][1]
    } else {
        int m = lane - 16;
        a0 = (m < 3) ? M[m * 4 + 2] : 0.0f;         // R[m][2]
        a1 = 0.0f;                                   // K=3 pad
    }
    // B operand (4x16 f32): V0 = {K=0 | K=2}, V1 = {K=1 | K=3}
    float b0, b1;
    if (lane < 16) {
        b0 = ((float)(w0 + lane) - CX) / FX;        // u  (K=0)
        b1 = ((float)h - CY) / FY;                  // v  (K=1)
    } else {
        b0 = 1.0f;                                   // K=2 row: ones
        b1 = 0.0f;                                   // K=3 pad
    }
    v2f A = {a0, a1};
    v2f B = {b0, b1};
    v8f C = {};
    v8f D = __builtin_amdgcn_wmma_f32_16x16x4_f32(false, A, false, B,
                                                  (short)0, C, false, false);
    if (lane < 16) {
        float x = D[0], y = D[1], z = D[2];
        float n = sqrtf(x * x + y * y + z * z);
        int pix = v * (H_IMG * W_IMG) + pix0 + lane;
        dirs[pix * 3 + 0] = x / n;
        dirs[pix * 3 + 1] = y / n;
        dirs[pix * 3 + 2] = z / n;
    }
}

// ===== Kernel 3: sphere trace + shade + per-block deterministic reduction =====
__global__ void k_raymarch(const float* __restrict__ sdf_rs,
                           const float* __restrict__ dirs,
                           const float* __restrict__ poses,
                           const float* __restrict__ rgb96,
                           const float* __restrict__ sem96,
                           const float* __restrict__ views,
                           float* __restrict__ partials) {
    const float OFFX[3] = {0.0f, 36.0f, 40.0f};
    const float OFFY[3] = {0.0f, 0.0f, 0.0f};
    const float OFFZ[3] = {0.0f, 44.0f, 23.0f};

    int ray = blockIdx.x * 256 + threadIdx.x;          // exactly 230400 rays
    int v = ray / (H_IMG * W_IMG);

    float dx = dirs[ray * 3 + 0];
    float dy = dirs[ray * 3 + 1];
    float dz = dirs[ray * 3 + 2];
    const float* M = poses + v * 16;
    float ox = M[0 * 4 + 3] / CAM_SCALE + OFFX[v];
    float oy = M[1 * 4 + 3] / CAM_SCALE + OFFY[v];
    float oz = M[2 * 4 + 3] / CAM_SCALE + OFFZ[v];

    float t = 0.0f;
    #pragma unroll 4
    for (int it = 0; it < N_STEPS; ++it) {
        float s = tri_sdf(sdf_rs, ox + t * dx, oy + t * dy, oz + t * dz);
        t += fmaxf(s, 0.25f) * 2.0f;
    }
    float px = ox + t * dx, py = oy + t * dy, pz = oz + t * dz;
    float s = tri_sdf(sdf_rs, px, py, pz);

    bool inside = (px >= 0.0f) && (px <= 159.0f) &&
                  (py >= 0.0f) && (py <= 159.0f) &&
                  (pz >= 0.0f) && (pz <= 159.0f);
    bool valid = inside && (fabsf(s) < 1.0f);

    float lsum = 0.0f, lcnt = 0.0f;
    if (valid) {
        // weights of 8 corners in the 160^3 space (same clamp as _trilerp)
        float cx = fminf(fmaxf(px, 0.0f), 158.999f);
        float cy = fminf(fmaxf(py, 0.0f), 158.999f);
        float cz = fminf(fmaxf(pz, 0.0f), 158.999f);
        int x0 = (int)cx, y0 = (int)cy, z0 = (int)cz;
        float fx = cx - (float)x0, fy = cy - (float)y0, fz = cz - (float)z0;
        int xs[2] = {x0, x0 + 1}, ys[2] = {y0, y0 + 1}, zs[2] = {z0, z0 + 1};
        float wxv[2] = {1.0f - fx, fx}, wyv[2] = {1.0f - fy, fy}, wzv[2] = {1.0f - fz, fz};

        float img0 = 0.0f, img1 = 0.0f, img2 = 0.0f, wsem = 0.0f;
        #pragma unroll
        for (int a = 0; a < 2; ++a)
        #pragma unroll
        for (int b = 0; b < 2; ++b)
        #pragma unroll
        for (int c = 0; c < 2; ++c) {
            float wt = wxv[a] * wyv[b] * wzv[c];
            // exact resized-rgb corner value = trilerp of 96^3 grid
            float sx = (float)xs[a] * SC;
            float sy = (float)ys[b] * SC;
            float sz = (float)zs[c] * SC;
            img0 += wt * tri96(rgb96 + 0 * (96 * 96 * 96), sx, sy, sz);
            img1 += wt * tri96(rgb96 + 1 * (96 * 96 * 96), sx, sy, sz);
            img2 += wt * tri96(rgb96 + 2 * (96 * 96 * 96), sx, sy, sz);
            // NN-resized sem corner: idx = floor(i * 96/160) = i*3/5
            int nx = (xs[a] * 3) / 5, ny = (ys[b] * 3) / 5, nz = (zs[c] * 3) / 5;
            wsem += wt * sem96[(nx * 96 + ny) * 96 + nz];
        }
        lsum = (fabsf(img0 - views[ray * 3 + 0]) +
                fabsf(img1 - views[ray * 3 + 1]) +
                fabsf(img2 - views[ray * 3 + 2])) * wsem;
        lcnt = 3.0f;
    }

    __shared__ float ssum[256];
    __shared__ float scnt[256];
    int tid = threadIdx.x;
    ssum[tid] = lsum;
    scnt[tid] = lcnt;
    __syncthreads();
    for (int off = 128; off > 0; off >>= 1) {
        if (tid < off) {
            ssum[tid] += ssum[tid + off];
            scnt[tid] += scnt[tid + off];
        }
        __syncthreads();
    }
    if (tid == 0) {
        partials[blockIdx.x * 2 + 0] = ssum[0];
        partials[blockIdx.x * 2 + 1] = scnt[0];
    }
}

// ===== Kernel 4: deterministic final reduction =====
__global__ void k_finalize(const float* __restrict__ partials,
                           float* __restrict__ out) {
    __shared__ float ssum[256];
    __shared__ float scnt[256];
    int tid = threadIdx.x;
    float s = 0.0f, c = 0.0f;
    for (int i = tid; i < RM_BLOCKS; i += 256) {
        s += partials[i * 2 + 0];
        c += partials[i * 2 + 1];
    }
    ssum[tid] = s;
    scnt[tid] = c;
    __syncthreads();
    for (int off = 128; off > 0; off >>= 1) {
        if (tid < off) {
            ssum[tid] += ssum[tid + off];
            scnt[tid] += scnt[tid + off];
        }
        __syncthreads();
    }
    if (tid == 0) {
        out[0] = 8.0f * ssum[0] / fmaxf(scnt[0], 1.0f);
    }
}

extern "C" void kernel_launch(void* const* d_in, const int* in_sizes, int n_in,
                              void* d_out, int out_size, void* d_ws, size_t ws_size,
                              hipStream_t stream) {
    const float* sdf96 = (const float*)d_in[0];
    const float* rgb96 = (const float*)d_in[1];
    const float* sem96 = (const float*)d_in[2];
    const float* poses = (const float*)d_in[3];
    const float* views = (const float*)d_in[4];
    float* out = (float*)d_out;

    char* ws = (char*)d_ws;
    float* sdf_rs   = (float*)(ws);                 // 160^3 floats  = 16,384,000 B
    float* dirs     = (float*)(ws + 16384000);      // 230400*3 f   =  2,764,800 B
    float* partials = (float*)(ws + 19148800);      // 1800 floats  =      7,200 B

    // 1) materialize the resized SDF (fits in L2 for the march)
    k_resize_sdf<<<(S_OUT * S_OUT * S_OUT) / 256, 256, 0, stream>>>(sdf96, sdf_rs);
    // 2) ray directions via WMMA (14400 strips == 1800 blocks * 8 waves exactly)
    k_dirs_wmma<<<1800, 256, 0, stream>>>(poses, dirs);
    // 3) sphere trace + shade + partial reduce (230400 rays == 900 * 256)
    k_raymarch<<<RM_BLOCKS, 256, 0, stream>>>(sdf_rs, dirs, poses, rgb96, sem96,
                                              views, partials);
    // 4) final deterministic reduce -> scalar
    k_finalize<<<1, 256, 0, stream>>>(partials, out);
}